// LIF_67714454389109
// MI455X (gfx1250) — compile-verified
//
#include <hip/hip_runtime.h>

// STBP LIF recurrence: h_t = (v + x_t)/2 ; s_t = (h_t >= 1) ; v = s_t ? 0 : h_t
// x [T=64, B=64, N=8192] f32 -> spikes, same shape.
// HBM-streaming kernel: 1 thread per 4 consecutive N-sites, v carried in
// registers over T. unroll 8 for ~16.8 MB loads-in-flight across the grid,
// NT b128 loads/stores, lane-masked global_prefetch_b8 16 steps ahead.

typedef __attribute__((ext_vector_type(4))) float v4f;

#define LIF_T 64

static __device__ __forceinline__ v4f lif_step(const v4f xt, v4f& v) {
    // h = v + (x - v)/tau, tau = 2, v_reset = 0  =>  h = 0.5*(v + x)
    const v4f h = (v + xt) * 0.5f;
    v4f s;
    s.x = (h.x >= 1.0f) ? 1.0f : 0.0f;
    s.y = (h.y >= 1.0f) ? 1.0f : 0.0f;
    s.z = (h.z >= 1.0f) ? 1.0f : 0.0f;
    s.w = (h.w >= 1.0f) ? 1.0f : 0.0f;
    v.x = (h.x >= 1.0f) ? 0.0f : h.x;
    v.y = (h.y >= 1.0f) ? 0.0f : h.y;
    v.z = (h.z >= 1.0f) ? 0.0f : h.z;
    v.w = (h.w >= 1.0f) ? 0.0f : h.w;
    return s;
}

__global__ __launch_bounds__(256) void lif_stbp_kernel(
    const v4f* __restrict__ x,   // in float4 units
    v4f* __restrict__ out,       // in float4 units
    int stride4)                 // (B*N)/4 : float4 stride between timesteps
{
    const size_t site = (size_t)blockIdx.x * blockDim.x + threadIdx.x;
    const v4f* __restrict__ xp = x + site;
    v4f* __restrict__ op = out + site;

    v4f v = {0.0f, 0.0f, 0.0f, 0.0f};

    // One prefetch lane per 128B cacheline (8 threads x 16B) to avoid
    // redundant GL2 prefetch requests.
    const bool pf_lane = ((threadIdx.x & 7u) == 0u);

    int t = 0;

    // Main loop: 48 steps with GL2 prefetch 16 timesteps ahead (in-bounds).
#pragma unroll 8
    for (; t < LIF_T - 16; ++t) {
        v4f xt = __builtin_nontemporal_load(xp + (size_t)t * stride4);
        if (pf_lane)
            __builtin_prefetch((const void*)(xp + (size_t)(t + 16) * stride4), 0, 1);
        v4f s = lif_step(xt, v);
        __builtin_nontemporal_store(s, op + (size_t)t * stride4);
    }

    // Tail: last 16 steps, no prefetch (would run past the tensor).
#pragma unroll 8
    for (; t < LIF_T; ++t) {
        v4f xt = __builtin_nontemporal_load(xp + (size_t)t * stride4);
        v4f s = lif_step(xt, v);
        __builtin_nontemporal_store(s, op + (size_t)t * stride4);
    }
}

extern "C" void kernel_launch(void* const* d_in, const int* in_sizes, int n_in,
                              void* d_out, int out_size, void* d_ws, size_t ws_size,
                              hipStream_t stream) {
    (void)n_in; (void)d_ws; (void)ws_size; (void)out_size;

    const float* x = (const float*)d_in[0];
    float* out = (float*)d_out;

    const int total = in_sizes[0];      // T * B * N = 33,554,432
    const int sites = total / LIF_T;    // B * N = 524,288  (multiple of 4)
    const int stride4 = sites >> 2;     // timestep stride in float4 units
    const int nthreads = sites >> 2;    // one thread per float4 of sites
    const int block = 256;
    const int grid = (nthreads + block - 1) / block;

    lif_stbp_kernel<<<grid, block, 0, stream>>>(
        (const v4f*)x, (v4f*)out, stride4);
}